// DMS_STGAT_32126355374744
// MI455X (gfx1250) — compile-verified
//
#include <hip/hip_runtime.h>
#include <math.h>

// ---------------------------------------------------------------------------
// DMS-STGAT attention fusion for MI455X (gfx1250, wave32).
//
// Algebraic collapse: h = x@W is only used through dot-products with the two
// halves of `a`, so u[i] = x[i]·(W@a0), v[i] = x[i]·(W@a1).  Spatial-PE and
// the sinusoid table are rank-1 terms folded into per-row scalars.  The
// remaining work is one streaming pass over src (5.12 MB) -> memory bound
// (~0.3us at 23.3 TB/s).  The per-row 4-wide dot batch is done with
// V_WMMA_F32_16X16X4_F32 (f32 WMMA, exact f32 accumulate) over 16-row tiles,
// K=128 in 32 steps.  LDS operands are zero-padded so the inner loop is a
// pure ds_load -> v_wmma stream with no per-lane masking VALU.
// ---------------------------------------------------------------------------

typedef __attribute__((ext_vector_type(2))) float v2f;
typedef __attribute__((ext_vector_type(8))) float v8f;

#define B_  16
#define C_  128
#define T_  25
#define J_  25

// workspace float offsets
#define WS_WA   0        // 4 x 128  : wa0_s, wa1_s, wa0_t, wa1_t
#define WS_SUM  512      // 2        : sum(wa0_s), sum(wa1_s)
#define WS_PU   514      // 25       : pos[t]·wa0_t
#define WS_PV   539      // 25       : pos[t]·wa1_t
#define WS_US   576      // 10000    : spatial u  (group-major [b*25+t][j])
#define WS_VS   10576    // 10000    : spatial v
#define WS_UT   20576    // 10000    : temporal u raw (group-major [b*25+j][t])
#define WS_VT   30576    // 10000    : temporal v raw

// ---------------------------------------------------------------------------
// Kernel 0: wa vectors, their sums, positional dot scalars.  Tiny (1 block).
// ---------------------------------------------------------------------------
__global__ __launch_bounds__(512)
void dms_prep_kernel(const float* __restrict__ Ws, const float* __restrict__ as_,
                     const float* __restrict__ Wt, const float* __restrict__ at_,
                     float* __restrict__ ws) {
  int tid = threadIdx.x;
  if (tid < 512) {
    int vec = tid >> 7, c = tid & 127;
    const float* W = (vec < 2) ? Ws : Wt;
    const float* a = (vec < 2) ? as_ : at_;
    int off = (vec & 1) * 256;
    float acc = 0.f;
    for (int f = 0; f < 256; ++f) acc += W[c * 256 + f] * a[off + f];
    ws[WS_WA + vec * 128 + c] = acc;
  }
  __syncthreads();
  if (tid < 2) {            // sums of the spatial wa vectors (for the PE term)
    float s = 0.f;
    for (int c = 0; c < 128; ++c) s += ws[WS_WA + tid * 128 + c];
    ws[WS_SUM + tid] = s;
  }
  if (tid < 50) {           // pu[t] = pos[t]·wa0_t ; pv[t] = pos[t]·wa1_t
    int sel = tid / 25, t = tid % 25;
    const float* wv = &ws[WS_WA + (2 + sel) * 128];
    float acc = 0.f;
    for (int c = 0; c < 128; ++c) {
      int k = c >> 1;
      double denom = pow(10000.0, (2.0 * (double)k) / 128.0);
      double ang = (double)t / denom;
      double sv = (c & 1) ? cos(ang) : sin(ang);
      float p = (float)sv * 1000.0f;   // table is cast f32 then *1000 (f32)
      acc += p * wv[c];
    }
    ws[(sel ? WS_PV : WS_PU) + t] = acc;
  }
}

// ---------------------------------------------------------------------------
// Kernel 1: one block per (b,t) group.  Loads the 25x128 slice of src once,
// computes the spatial-PE scalars, then the 4 dots per row via f32 WMMA
// (two 16-row tiles, one per wave; B columns 0..3 = wa vectors, 4..15 = 0).
// xs rows 25..31 and lwa rows 4..15 are zeroed so no lane masking is needed.
// ---------------------------------------------------------------------------
__global__ __launch_bounds__(64)
void dms_stage1_kernel(const float* __restrict__ src, float* __restrict__ ws) {
  __shared__ float xs[32][130];     // row-major tile, padded stride, padded rows
  __shared__ float lwa[16][128];    // rows 0..3 = wa vectors, 4..15 = zero
  __shared__ float pe[25];

  int g = blockIdx.x;               // g = b*25 + t
  int b = g / 25, t = g % 25;
  int tid = threadIdx.x;

  // zero the padding (rows 25..31 of xs, rows 4..15 of lwa)
  for (int i = tid; i < 7 * 130; i += 64) xs[25 + i / 130][i % 130] = 0.f;
  for (int i = tid; i < 12 * 128; i += 64) lwa[4 + (i >> 7)][i & 127] = 0.f;
  for (int i = tid; i < 512; i += 64) lwa[i >> 7][i & 127] = ws[WS_WA + i];

  // src[b][c][t][j]: for fixed (b,t), element (j,c) at c*625 + j
  const float* sp = src + (size_t)b * C_ * T_ * J_ + (size_t)t * J_;
  for (int i = tid; i < 25 * 128; i += 64) {
    int c = i / 25, j = i % 25;
    xs[j][c] = sp[(size_t)c * T_ * J_ + j];
  }
  __syncthreads();

  if (tid < 25) {                   // exp(-||x_j/1000 - x_8/1000||)
    float acc = 0.f;
    for (int c = 0; c < 128; ++c) {
      float d = xs[tid][c] / 1000.0f - xs[8][c] / 1000.0f;
      acc += d * d;
    }
    pe[tid] = expf(-sqrtf(acc));
  }
  __syncthreads();

  int lane = tid & 31;
  int w    = tid >> 5;              // wave id: tile of rows 16w..16w+15
  int n    = lane & 15;             // D column / B column for this lane
  int hi   = lane >> 4;             // K-half (A/B: lanes 16-31 hold K+2)
  int arow = w * 16 + n;            // A row held by this lane (zero-padded)

  v8f acc = {};
#pragma unroll
  for (int kk = 0; kk < 32; ++kk) {
    int cb = 4 * kk + 2 * hi;
    v2f a, bb;
    a.x = xs[arow][cb];  a.y = xs[arow][cb + 1];
    bb.x = lwa[n][cb];   bb.y = lwa[n][cb + 1];
    acc = __builtin_amdgcn_wmma_f32_16x16x4_f32(
        /*neg_a=*/false, a, /*neg_b=*/false, bb,
        /*c_mod=*/(short)0, acc, /*reuse_a=*/false, /*reuse_b=*/false);
  }

  // D layout: VGPR r, lane L -> (M = r + 8*(L>>4), N = L&15)
  if (n < 4) {
    float s0 = ws[WS_SUM + 0], s1 = ws[WS_SUM + 1];
#pragma unroll
    for (int r = 0; r < 8; ++r) {
      int m = w * 16 + r + 8 * hi;
      if (m < 25) {
        float d = acc[r];
        int j = m;
        if (n == 0)      ws[WS_US + g * 25 + j] = d + pe[j] * s0;
        else if (n == 1) ws[WS_VS + g * 25 + j] = d + pe[j] * s1;
        else if (n == 2) ws[WS_UT + (b * 25 + j) * 25 + t] = d;   // transpose to (b,j) groups
        else             ws[WS_VT + (b * 25 + j) * 25 + t] = d;
      }
    }
  }
}

// ---------------------------------------------------------------------------
// Kernel 2: one wave per group (400 spatial + 400 temporal).  Builds the
// 25x25 logit matrix from u/v via the reference's reshape row-mapping,
// leaky-relu, then double softmax over p, and writes the 625 outputs.
// ---------------------------------------------------------------------------
__global__ __launch_bounds__(32)
void dms_stage2_kernel(const float* __restrict__ ws, float* __restrict__ out) {
  __shared__ float u[25], v[25];
  __shared__ float em[25][26];

  int g = blockIdx.x;
  int tid = threadIdx.x;
  bool temporal = (g >= 400);
  int gg = temporal ? g - 400 : g;

  if (tid < 25) {
    if (!temporal) {
      u[tid] = ws[WS_US + gg * 25 + tid];
      v[tid] = ws[WS_VS + gg * 25 + tid];
    } else {
      u[tid] = ws[WS_UT + gg * 25 + tid] + ws[WS_PU + tid];
      v[tid] = ws[WS_VT + gg * 25 + tid] + ws[WS_PV + tid];
    }
  }
  __syncthreads();

  if (tid < 25) {
    int q = tid;
    float col[25];
    float mx = -INFINITY;
#pragma unroll
    for (int p = 0; p < 25; ++p) {
      int m0 = 50 * p + 2 * q;
      int r0 = (m0 < 625) ? (m0 / 25) : ((m0 - 625) % 25);
      int m1 = m0 + 1;
      int r1 = (m1 < 625) ? (m1 / 25) : ((m1 - 625) % 25);
      float e = u[r0] + v[r1];
      e = (e >= 0.f) ? e : 0.2f * e;          // leaky_relu, slope 0.2
      col[p] = e;
      mx = fmaxf(mx, e);
    }
    float sum = 0.f;
#pragma unroll
    for (int p = 0; p < 25; ++p) { col[p] = expf(col[p] - mx); sum += col[p]; }
    float mx2 = -INFINITY;
#pragma unroll
    for (int p = 0; p < 25; ++p) { col[p] = col[p] / sum; mx2 = fmaxf(mx2, col[p]); }
    float sum2 = 0.f;
#pragma unroll
    for (int p = 0; p < 25; ++p) { col[p] = expf(col[p] - mx2); sum2 += col[p]; }
#pragma unroll
    for (int p = 0; p < 25; ++p) em[p][q] = col[p] / sum2;
  }
  __syncthreads();

  float* ob = out + (temporal ? 250000 : 0) + gg * 625;
  for (int i = tid; i < 625; i += 32) ob[i] = em[i / 25][i % 25];
}

// ---------------------------------------------------------------------------
extern "C" void kernel_launch(void* const* d_in, const int* in_sizes, int n_in,
                              void* d_out, int out_size, void* d_ws, size_t ws_size,
                              hipStream_t stream) {
  const float* src = (const float*)d_in[0];   // (16,128,25,25)
  const float* Ws  = (const float*)d_in[1];   // (128,256)
  const float* a_s = (const float*)d_in[2];   // (512,1)
  const float* Wt  = (const float*)d_in[3];   // (128,256)
  const float* a_t = (const float*)d_in[4];   // (512,1)
  float* out = (float*)d_out;                 // 500000 floats
  float* ws  = (float*)d_ws;                  // needs ~162 KB

  dms_prep_kernel<<<1, 512, 0, stream>>>(Ws, a_s, Wt, a_t, ws);
  dms_stage1_kernel<<<400, 64, 0, stream>>>(src, ws);
  dms_stage2_kernel<<<800, 32, 0, stream>>>(ws, out);
}